// SimOTAAssigner_66829691125889
// MI455X (gfx1250) — compile-verified
//
#include <hip/hip_runtime.h>
#include <hip/hip_bf16.h>

typedef float v2f __attribute__((ext_vector_type(2)));
typedef float v8f __attribute__((ext_vector_type(8)));
typedef unsigned int u32x4 __attribute__((ext_vector_type(4)));
typedef int i32x4 __attribute__((ext_vector_type(4)));
typedef int i32x8 __attribute__((ext_vector_type(8)));

#define IOU_W 3.0f

__device__ __forceinline__ unsigned orderbits(float f) {
    unsigned u = __float_as_uint(f);
    return (u & 0x80000000u) ? ~u : (u | 0x80000000u);
}

// ---------------------------------------------------------------------------
// Kernel A: fused pair-score (WMMA one-hot matmul) + IoU + cost epilogue.
// One wave computes a 16(gt) x 16(anchor) tile of cost[G,N] & ious_masked[G,N].
// GT boxes staged block-wide into LDS via the Tensor Data Mover.
// ---------------------------------------------------------------------------
__global__ void simota_cost_kernel(const float* __restrict__ pd_scores,
                                   const float* __restrict__ pd_bboxes,
                                   const float* __restrict__ anc_points,
                                   const int* __restrict__ gt_labels, int lstride,
                                   const float* __restrict__ gt_bboxes,
                                   float* __restrict__ cost,
                                   float* __restrict__ ioum,
                                   int N, int G, int C) {
    __shared__ float s_gtb[256 * 4];   // gt boxes (TDM destination)
    __shared__ int   s_cls[256];

    const int tid = threadIdx.x;

    // ---- TDM: DMA gt_bboxes (G*4 f32, 1-D tensor) into LDS ----
    {
        unsigned lds_off = (unsigned)(unsigned long long)(void*)&s_gtb[0];
        unsigned long long ga = (unsigned long long)gt_bboxes;
        unsigned ne = (unsigned)(G * 4);          // elements (4B each)
        u32x4 g0;
        g0.x = 1u;                                 // count=1, user mode
        g0.y = lds_off;                            // lds_addr
        g0.z = (unsigned)(ga & 0xFFFFFFFFu);       // global_addr lo
        g0.w = (unsigned)((ga >> 32) & 0x1FFFFFFu) | (2u << 30); // addr hi | type=2
        i32x8 g1;
        g1[0] = (int)(2u << 16);                   // data_size = 4 bytes
        g1[1] = (int)((ne & 0xFFFFu) << 16);       // tensor_dim0[15:0]
        g1[2] = (int)(((ne >> 16) & 0xFFFFu) | (1u << 16)); // dim0 hi | tensor_dim1=1
        g1[3] = (int)((ne & 0xFFFFu) << 16);       // tile_dim0 = ne
        g1[4] = 0;                                 // tile_dim1/2 unused
        g1[5] = (int)ne;                           // tensor_dim0_stride lo32
        g1[6] = 0;
        g1[7] = 0;
        i32x4 zz4 = {0, 0, 0, 0};
        i32x8 zz8 = {0, 0, 0, 0, 0, 0, 0, 0};
        if (tid == 0)
            __builtin_amdgcn_tensor_load_to_lds(g0, g1, zz4, zz4, zz8, 0);
        __builtin_amdgcn_s_wait_tensorcnt(0);
    }
    for (int i = tid; i < G; i += blockDim.x)
        s_cls[i] = gt_labels[i * lstride];
    __syncthreads();

    const int wave = tid >> 5, lane = tid & 31;
    const int ntiles = N >> 4;
    const int atile = blockIdx.x * (blockDim.x >> 5) + wave;
    if (atile >= ntiles) return;

    const int g0r = blockIdx.y * 16;
    const int a0  = atile * 16;
    const int nidx  = lane & 15;
    const int khalf = (lane >> 4) << 1;   // 0 or 2 (K sub-pair per half-wave)
    const int a = a0 + nidx;

    // A one-hot row for this lane: M = gt (g0r + nidx)
    const int gA   = g0r + nidx;
    const int clsA = (gA < G) ? s_cls[gA] : -1;

    v8f acc = {0.f, 0.f, 0.f, 0.f, 0.f, 0.f, 0.f, 0.f};
    const float* srow = pd_scores + (size_t)a * C;
    for (int k0 = 0; k0 < C; k0 += 4) {
        const int ka = k0 + khalf;
        v2f av, bv;
        av.x = (clsA == ka)     ? 1.0f : 0.0f;
        av.y = (clsA == ka + 1) ? 1.0f : 0.0f;
        const float2 bs = *reinterpret_cast<const float2*>(srow + ka);
        bv.x = bs.x; bv.y = bs.y;
        acc = __builtin_amdgcn_wmma_f32_16x16x4_f32(false, av, false, bv,
                                                    (short)0, acc, false, false);
    }

    // epilogue: per-lane anchor data, per-VGPR gt row
    const float4 ab = *reinterpret_cast<const float4*>(pd_bboxes + (size_t)a * 4);
    const float2 ap = *reinterpret_cast<const float2*>(anc_points + (size_t)a * 2);
    const float areaA = (ab.z - ab.x) * (ab.w - ab.y);
    const int ghalf = (lane >> 4) << 3;   // 0 or 8

    for (int r = 0; r < 8; ++r) {
        const int g = g0r + r + ghalf;
        if (g >= G) continue;
        const float gx0 = s_gtb[g * 4 + 0], gy0 = s_gtb[g * 4 + 1];
        const float gx1 = s_gtb[g * 4 + 2], gy1 = s_gtb[g * 4 + 3];
        // IoU
        const float ix = fminf(gx1, ab.z) - fmaxf(gx0, ab.x);
        const float iy = fminf(gy1, ab.w) - fmaxf(gy0, ab.y);
        const float ia = fmaxf(ix, 0.f) * fmaxf(iy, 0.f);
        const float uni = (gx1 - gx0) * (gy1 - gy0) + areaA - ia + 1e-9f;
        const float iou = ia / uni;
        // in-center
        const float dmin = fminf(fminf(ap.x - gx0, ap.y - gy0),
                                 fminf(gx1 - ap.x, gy1 - ap.y));
        const bool inc = dmin > 0.01f;
        // cost
        const float s = acc[r];
        const float sp = fmaxf(-s, 0.f) + log1pf(__expf(-fabsf(s)));
        const float ci = -__logf(iou + 1e-8f);
        const float cst = sp + IOU_W * ci + (inc ? 0.f : 1e5f);
        const size_t off = (size_t)g * N + a;
        cost[off] = cst;
        ioum[off] = inc ? iou : 0.f;
    }
}

// ---------------------------------------------------------------------------
// Kernel B: dynamic_ks[g] = clamp((int)sum_n ious_masked[g,n], 1, ...)
// ---------------------------------------------------------------------------
__global__ void simota_rowsum_kernel(const float* __restrict__ ioum,
                                     int* __restrict__ dks, int N) {
    const int g = blockIdx.x;
    __shared__ float red[256];
    float s = 0.f;
    for (int i = threadIdx.x; i < N; i += blockDim.x)
        s += ioum[(size_t)g * N + i];
    red[threadIdx.x] = s;
    __syncthreads();
    for (int w = 128; w > 0; w >>= 1) {
        if ((int)threadIdx.x < w) red[threadIdx.x] += red[threadIdx.x + w];
        __syncthreads();
    }
    if (threadIdx.x == 0) {
        int k = (int)red[0];
        dks[g] = (k < 1) ? 1 : k;
    }
}

// ---------------------------------------------------------------------------
// Kernel C: per-row exact k-th smallest cost via 4-pass 8-bit radix select.
// thrbits[g] = order-transformed bits of the k-th smallest value.
// ---------------------------------------------------------------------------
__global__ void simota_kselect_kernel(const float* __restrict__ cost,
                                      const int* __restrict__ dks,
                                      unsigned* __restrict__ thrbits, int N) {
    const int g = blockIdx.x;
    const float* row = cost + (size_t)g * N;
    __shared__ int hist[256];
    __shared__ unsigned s_prefix;
    __shared__ int s_kk;
    if (threadIdx.x == 0) { s_prefix = 0u; s_kk = dks[g]; }
    __syncthreads();
    for (int shift = 24; shift >= 0; shift -= 8) {
        for (int i = threadIdx.x; i < 256; i += blockDim.x) hist[i] = 0;
        __syncthreads();
        const unsigned prefix = s_prefix;
        const unsigned mask = (shift == 24) ? 0u : (0xFFFFFFFFu << (shift + 8));
        for (int i = threadIdx.x; i < N; i += blockDim.x) {
            const unsigned u = orderbits(row[i]);
            if ((u & mask) == prefix) atomicAdd(&hist[(u >> shift) & 255], 1);
        }
        __syncthreads();
        if (threadIdx.x == 0) {
            int kk = s_kk, acc = 0, b = 0;
            for (; b < 255; ++b) {
                if (acc + hist[b] >= kk) break;
                acc += hist[b];
            }
            s_kk = kk - acc;
            s_prefix = prefix | ((unsigned)b << shift);
        }
        __syncthreads();
    }
    if (threadIdx.x == 0) thrbits[g] = s_prefix;
}

// ---------------------------------------------------------------------------
// Kernel D: per-anchor match count, conflict resolution, labels/bboxes/fg.
// ---------------------------------------------------------------------------
__global__ void simota_assign_kernel(const float* __restrict__ cost,
                                     const float* __restrict__ ioum,
                                     const unsigned* __restrict__ thrbits,
                                     const int* __restrict__ gt_labels, int lstride,
                                     const float* __restrict__ gt_bboxes,
                                     float* __restrict__ out,
                                     int* __restrict__ lbl_i,
                                     float* __restrict__ miou_w,
                                     int N, int G, int C) {
    const int n = blockIdx.x * blockDim.x + threadIdx.x;
    if (n >= N) return;
    int count = 0, first = -1, amin = 0;
    float cmin = 3.4e38f;
    for (int g = 0; g < G; ++g) {
        const float c = cost[(size_t)g * N + n];
        const unsigned u = orderbits(c);
        if (u <= thrbits[g]) { count++; if (first < 0) first = g; }
        if (c < cmin) { cmin = c; amin = g; }
    }
    const bool fg = count > 0;
    const int matched = (count > 1) ? amin : (first >= 0 ? first : 0);
    const int lbl = fg ? gt_labels[matched * lstride] : C;
    const float mi = fg ? ioum[(size_t)matched * N + n] : 0.f;

    out[n] = (float)lbl;                                   // assigned_labels
    float4 bb = make_float4(0.f, 0.f, 0.f, 0.f);
    if (fg) bb = *reinterpret_cast<const float4*>(gt_bboxes + (size_t)matched * 4);
    float* ob = out + (size_t)N + (size_t)n * 4;           // assigned_bboxes
    ob[0] = bb.x; ob[1] = bb.y; ob[2] = bb.z; ob[3] = bb.w;
    out[(size_t)N * 85 + n] = fg ? 1.f : 0.f;              // fg_mask
    lbl_i[n] = lbl;
    miou_w[n] = mi;
}

// ---------------------------------------------------------------------------
// Kernel E: assigned_scores[n,c] = (c == label[n]) * matched_iou[n]
// ---------------------------------------------------------------------------
__global__ void simota_scores_kernel(const int* __restrict__ lbl_i,
                                     const float* __restrict__ miou_w,
                                     float* __restrict__ out_scores,
                                     int N, int C) {
    const size_t idx = (size_t)blockIdx.x * blockDim.x + threadIdx.x;
    if (idx >= (size_t)N * C) return;
    const int n = (int)(idx / C);
    const int c = (int)(idx % C);
    out_scores[idx] = (c == lbl_i[n]) ? miou_w[n] : 0.f;
}

extern "C" void kernel_launch(void* const* d_in, const int* in_sizes, int n_in,
                              void* d_out, int out_size, void* d_ws, size_t ws_size,
                              hipStream_t stream) {
    const float* pd_scores  = (const float*)d_in[0];
    const float* pd_bboxes  = (const float*)d_in[1];
    const float* anc_points = (const float*)d_in[2];
    const int*   gt_labels  = (const int*)d_in[3];
    const float* gt_bboxes  = (const float*)d_in[4];

    const int N = in_sizes[2] / 2;                 // anc_points [N,2]
    const int G = in_sizes[4] / 4;                 // gt_bboxes [G,4]
    const int C = in_sizes[0] / N;                 // pd_scores [N,C]
    int lstride = (G > 0) ? in_sizes[3] / G : 1;   // 1 (int32) or 2 (int64 lo-dword)
    if (lstride < 1) lstride = 1;

    float* out = (float*)d_out;

    // workspace layout
    const size_t GN = (size_t)G * N;
    float*    cost = (float*)d_ws;
    float*    ioum = cost + GN;
    int*      dks  = (int*)(ioum + GN);
    unsigned* thr  = (unsigned*)(dks + G);
    int*      lbli = (int*)(thr + G);
    float*    miou = (float*)(lbli + N);

    // A: cost + masked IoU (WMMA + TDM)
    {
        const int wavesPerBlock = 8;
        const int ntiles = (N + 15) / 16;
        dim3 grid((ntiles + wavesPerBlock - 1) / wavesPerBlock, (G + 15) / 16);
        simota_cost_kernel<<<grid, dim3(32 * wavesPerBlock), 0, stream>>>(
            pd_scores, pd_bboxes, anc_points, gt_labels, lstride, gt_bboxes,
            cost, ioum, N, G, C);
    }
    // B: dynamic_ks
    simota_rowsum_kernel<<<G, 256, 0, stream>>>(ioum, dks, N);
    // C: per-row k-th smallest threshold
    simota_kselect_kernel<<<G, 256, 0, stream>>>(cost, dks, thr, N);
    // D: per-anchor assignment + conflict resolution
    simota_assign_kernel<<<(N + 255) / 256, 256, 0, stream>>>(
        cost, ioum, thr, gt_labels, lstride, gt_bboxes, out, lbli, miou, N, G, C);
    // E: one-hot score fill
    {
        const size_t tot = (size_t)N * C;
        simota_scores_kernel<<<(unsigned)((tot + 255) / 256), 256, 0, stream>>>(
            lbli, miou, out + (size_t)N * 5, N, C);
    }
}